// SimpleGCN_83262236000846
// MI455X (gfx1250) — compile-verified
//
#include <hip/hip_runtime.h>

typedef __attribute__((ext_vector_type(2))) float v2f;
typedef __attribute__((ext_vector_type(8))) float v8f;

#define F_IN  128
#define F_HID 64

// ---------------- init: deg=1 (self loop), hagg=0, acc=0 ----------------
__global__ void k_init(float* __restrict__ deg, float* __restrict__ hagg,
                       float* __restrict__ acc, int N) {
  long long tid = (long long)blockIdx.x * blockDim.x + threadIdx.x;
  long long nt  = (long long)gridDim.x * blockDim.x;
  long long tot = (long long)N * F_HID;
  for (long long i = tid; i < tot; i += nt) hagg[i] = 0.0f;
  for (long long i = tid; i < N;   i += nt) deg[i]  = 1.0f;  // self loop
  if (tid == 0) acc[0] = 0.0f;
}

// ---------------- degree over targets (col) ----------------
__global__ void k_degree(const long long* __restrict__ col, float* __restrict__ deg,
                         long long E) {
  long long tid = (long long)blockIdx.x * blockDim.x + threadIdx.x;
  long long nt  = (long long)gridDim.x * blockDim.x;
  for (long long e = tid; e < E; e += nt)
    atomicAdd(&deg[(int)col[e]], 1.0f);
}

// ---------------- dinv = rsqrt(deg) (deg >= 1 always) ----------------
__global__ void k_dinv(float* __restrict__ deg, int N) {
  long long tid = (long long)blockIdx.x * blockDim.x + threadIdx.x;
  long long nt  = (long long)gridDim.x * blockDim.x;
  for (long long i = tid; i < N; i += nt)
    deg[i] = rsqrtf(deg[i]);
}

// ---------------- h1 = x @ W1  via V_WMMA_F32_16X16X4_F32 ----------------
// One wave computes one 16x16 output tile; block = (32,4) => 4 N-tiles cover H=64.
// A (16x4 f32): lane l: row = l&15, ks = (l>>4)*2, a = {x[row,k+ks], x[row,k+ks+1]}
// B (4x16 f32): lane l: col = l&15, ks = (l>>4)*2, b = {W[k+ks,col], W[k+ks+1,col]}
// C/D (16x16 f32, 8 VGPR): lane l: col = l&15, rows = (l>>4)*8 + v
__global__ __launch_bounds__(128) void k_gemm1(const float* __restrict__ x,
                                               const float* __restrict__ W,
                                               float* __restrict__ h, int N) {
  const int lane = threadIdx.x & 31;
  const int m0   = blockIdx.x * 16;
  const int n0   = (int)threadIdx.y * 16;
  const int mr   = lane & 15;
  const int ks   = (lane >> 4) * 2;
  const int nc   = lane & 15;

  int row = m0 + mr; if (row >= N) row = N - 1;       // clamp: EXEC stays all-ones
  const float* xrow = x + (long long)row * F_IN;

#if __has_builtin(__builtin_amdgcn_wmma_f32_16x16x4_f32)
  v8f acc = {};
  for (int k = 0; k < F_IN; k += 4) {
    v2f a, b;
    a[0] = xrow[k + ks];
    a[1] = xrow[k + ks + 1];
    b[0] = W[(long long)(k + ks)     * F_HID + n0 + nc];
    b[1] = W[(long long)(k + ks + 1) * F_HID + n0 + nc];
    acc = __builtin_amdgcn_wmma_f32_16x16x4_f32(false, a, false, b,
                                                (short)0, acc, false, false);
  }
  const int mb = (lane >> 4) * 8;
  float* hp = h + (long long)(m0 + mb) * F_HID + n0 + nc;
  if (m0 + 16 <= N) {
    // wave-uniform full-tile fast path: 8 unguarded coalesced stores
    #pragma unroll
    for (int v = 0; v < 8; ++v) hp[(long long)v * F_HID] = acc[v];
  } else {
    // ragged tail tile (never taken when N % 16 == 0)
    #pragma unroll
    for (int v = 0; v < 8; ++v)
      if (m0 + mb + v < N) hp[(long long)v * F_HID] = acc[v];
  }
#else
  // Fallback: plain VALU FMA (signals wmma builtin missing via histogram)
  for (int n = 0; n < 16; ++n) {
    float s = 0.0f;
    for (int k = 0; k < F_IN; ++k) s += xrow[k] * W[(long long)k * F_HID + n0 + n];
    if (lane < 16 && threadIdx.y == 0 && m0 + mr < N)
      h[(long long)(m0 + mr) * F_HID + n0 + n] = s;
  }
#endif
}

// ---------------- layer-1 edge scatter: hagg[col] += norm * h1[row] ----------------
// One wave per edge; 64 floats = one float2 per lane. Working set (h1 + hagg +
// edge index ~ 103 MB) fits in the 192 MB L2, so atomics resolve at L2.
__global__ void k_edge1(const long long* __restrict__ row,
                        const long long* __restrict__ col,
                        const float* __restrict__ dinv,
                        const float* __restrict__ h1,
                        float* __restrict__ hagg, long long E) {
  const int lane = threadIdx.x & 31;
  long long gw = (long long)blockIdx.x * (blockDim.x >> 5) + (threadIdx.x >> 5);
  long long nw = (long long)gridDim.x * (blockDim.x >> 5);
  for (long long e = gw; e < E; e += nw) {
    int r = (int)row[e];
    int c = (int)col[e];
    float norm = dinv[r] * dinv[c];
    float2 v = ((const float2*)(h1 + (long long)r * F_HID))[lane];
    float* dst = hagg + (long long)c * F_HID + 2 * lane;
    atomicAdd(dst,     norm * v.x);
    atomicAdd(dst + 1, norm * v.y);
  }
}

// ---------------- self-loop + bias + ReLU (in place on hagg) ----------------
__global__ void k_bias_relu(float* __restrict__ hagg, const float* __restrict__ h1,
                            const float* __restrict__ dinv, const float* __restrict__ b1,
                            int N) {
  long long tid = (long long)blockIdx.x * blockDim.x + threadIdx.x;
  long long nt  = (long long)gridDim.x * blockDim.x;
  long long tot = (long long)N * F_HID;
  for (long long i = tid; i < tot; i += nt) {
    int node = (int)(i >> 6);
    int f    = (int)(i & 63);
    float d  = dinv[node];
    float v  = hagg[i] + d * d * h1[i] + b1[f];
    hagg[i]  = fmaxf(v, 0.0f);
  }
}

// ---------------- h2 = relu(h1agg) @ W2 (64-wide matvec, wave per node) ----------------
__global__ void k_h2(const float* __restrict__ hrelu, const float* __restrict__ W2,
                     float* __restrict__ h2, int N) {
  const int lane = threadIdx.x & 31;
  long long gw = (long long)blockIdx.x * (blockDim.x >> 5) + (threadIdx.x >> 5);
  long long nw = (long long)gridDim.x * (blockDim.x >> 5);
  for (long long i = gw; i < N; i += nw) {
    const float* hr = hrelu + i * F_HID;
    float s = hr[lane] * W2[lane] + hr[lane + 32] * W2[lane + 32];
    #pragma unroll
    for (int o = 16; o > 0; o >>= 1) s += __shfl_xor(s, o, 32);
    if (lane == 0) h2[i] = s;
  }
}

// ---------------- final reduction: sum over edges + self loops ----------------
__global__ void k_final(const long long* __restrict__ row,
                        const long long* __restrict__ col,
                        const float* __restrict__ dinv,
                        const float* __restrict__ h2,
                        float* __restrict__ acc, long long E, int N) {
  long long tid = (long long)blockIdx.x * blockDim.x + threadIdx.x;
  long long nt  = (long long)gridDim.x * blockDim.x;
  float s = 0.0f;
  for (long long e = tid; e < E; e += nt) {
    int r = (int)row[e];
    int c = (int)col[e];
    s += dinv[r] * dinv[c] * h2[r];
  }
  for (long long i = tid; i < N; i += nt) {
    float d = dinv[i];
    s += d * d * h2[i];
  }
  #pragma unroll
  for (int o = 16; o > 0; o >>= 1) s += __shfl_xor(s, o, 32);
  if ((threadIdx.x & 31) == 0) atomicAdd(acc, s);
}

// ---------------- write output: mean + b2 ----------------
__global__ void k_out(const float* __restrict__ acc, const float* __restrict__ b2,
                      float* __restrict__ out, int N) {
  if (threadIdx.x == 0 && blockIdx.x == 0)
    out[0] = acc[0] / (float)N + b2[0];
}

extern "C" void kernel_launch(void* const* d_in, const int* in_sizes, int n_in,
                              void* d_out, int out_size, void* d_ws, size_t ws_size,
                              hipStream_t stream) {
  const float*     x  = (const float*)d_in[0];
  const long long* ei = (const long long*)d_in[1];   // [2, E] int64
  const float*     W1 = (const float*)d_in[2];
  const float*     b1 = (const float*)d_in[3];
  const float*     W2 = (const float*)d_in[4];
  const float*     b2 = (const float*)d_in[5];
  float*           out = (float*)d_out;

  const int       N = in_sizes[0] / F_IN;
  const long long E = (long long)in_sizes[1] / 2;
  const long long* row = ei;        // sources
  const long long* col = ei + E;    // targets

  // workspace layout (floats)
  float* ws   = (float*)d_ws;
  float* deg  = ws;                              // N   (becomes dinv)
  float* h1   = deg  + N;                        // N*64
  float* hagg = h1   + (long long)N * F_HID;     // N*64 (becomes relu'd hidden)
  float* h2   = hagg + (long long)N * F_HID;     // N
  float* acc  = h2   + N;                        // 1

  k_init  <<<2048, 256, 0, stream>>>(deg, hagg, acc, N);
  k_degree<<<8192, 256, 0, stream>>>(col, deg, E);
  k_dinv  <<<1024, 256, 0, stream>>>(deg, N);

  dim3 gblk(32, 4);
  int mtiles = (N + 15) / 16;
  k_gemm1 <<<mtiles, gblk, 0, stream>>>(x, W1, h1, N);

  k_edge1    <<<16384, 256, 0, stream>>>(row, col, deg, h1, hagg, E);
  k_bias_relu<<<4096, 256, 0, stream>>>(hagg, h1, deg, b1, N);
  k_h2       <<<8192, 256, 0, stream>>>(hagg, W2, h2, N);
  k_final    <<<4096, 256, 0, stream>>>(row, col, deg, h2, acc, E, N);
  k_out      <<<1, 32, 0, stream>>>(acc, b2, out, N);
}